// CTCLayer_87892210745816
// MI455X (gfx1250) — compile-verified
//
#include <hip/hip_runtime.h>
#include <stdint.h>

#define NEGV  (-1e30f)
#define EPSV  (1e-7f)

#define B_     128
#define T_     1024
#define C_     256
#define ROWB   (C_ * 4)       // 1024 bytes per class row
#define LMAX_  64
#define S_     129            // 2*LMAX+1
#define DEPTH  16             // async rows in flight
#define NBUF   18             // DEPTH+2: lets a single barrier per step cover the WAR on the ring
#define BLOCK  160            // 5 waves; states live in lanes 0..128

// CDNA5 async global->LDS, GVS addressing: SGPR64 base + per-lane u32 byte offset.
__device__ __forceinline__ void async_b128_to_lds(uint32_t lds_addr, uint32_t voff, uint64_t sbase) {
    asm volatile("global_load_async_to_lds_b128 %0, %1, %2"
                 :: "v"(lds_addr), "v"(voff), "s"(sbase)
                 : "memory");
}

__device__ __forceinline__ void wait_async_le(int steady) {
    if (steady)
        asm volatile("s_wait_asynccnt 15" ::: "memory");   // keep DEPTH-1 in flight
    else
        asm volatile("s_wait_asynccnt 0"  ::: "memory");   // drain tail
}

__global__ __launch_bounds__(BLOCK)
void ctc_loss_kernel(const int*   __restrict__ y_true,
                     const float* __restrict__ y_pred,
                     const int*   __restrict__ input_length,
                     const int*   __restrict__ label_length,
                     float*       __restrict__ out)
{
    __shared__ __align__(16) float rowbuf[NBUF][C_];   // 18 KB ring of class rows
    __shared__ float alphaA[S_ + 3];
    __shared__ float alphaB[S_ + 3];
    __shared__ int   labels[LMAX_];

    const int b    = blockIdx.x;
    const int tid  = threadIdx.x;
    const int L    = label_length[b];
    const int Tlen = input_length[b];
    const uint64_t gbase = (uint64_t)(uintptr_t)(y_pred + (size_t)b * T_ * C_);
    const uint32_t rb0   = (uint32_t)(uintptr_t)(void*)&rowbuf[0][0];  // LDS byte offset of ring

    if (tid < LMAX_) labels[tid] = y_true[b * LMAX_ + tid];

    const uint32_t lane_byte = (uint32_t)tid * 16u;      // 16 B per lane (b128)

    // Prime the pipeline: rows 0..DEPTH-1 (waves 0,1: 64 lanes x 16B = 1KB/row).
    if (tid < 64) {
        #pragma unroll
        for (int i = 0; i < DEPTH; ++i)
            async_b128_to_lds(rb0 + (uint32_t)i * ROWB + lane_byte,
                              (uint32_t)i * ROWB + lane_byte, gbase);
    }
    __syncthreads();   // labels visible

    // Per-state constants (registers)
    int  ext_s   = C_ - 1;         // blank
    bool skip_s  = false;
    bool valid_s = false;
    if (tid < S_) {
        if (tid & 1) {
            const int li = tid >> 1;
            ext_s  = labels[li];
            skip_s = (tid >= 3) && (labels[li] != labels[li - 1]);
        }
        valid_s = tid < (2 * L + 1);
    }

    float* acur = alphaA;
    float* anxt = alphaB;

    int      rslot = 0;                       // ring slot of row t
    int      pslot = DEPTH % NBUF;            // ring slot of row t+DEPTH
    uint32_t pvoff = (uint32_t)DEPTH * ROWB + lane_byte;   // global byte offset of row t+DEPTH

    for (int t = 0; t < T_; ++t) {
        const int steady = (t + DEPTH < T_);
        if (tid < 64) {
            if (steady)
                async_b128_to_lds(rb0 + (uint32_t)pslot * ROWB + lane_byte, pvoff, gbase);
            wait_async_le(steady);            // row t resident for waves 0,1
        }
        __syncthreads();  // single barrier/step: publishes row t AND last step's alpha writes;
                          // also orders compute(t-2)'s ring reads before this step's issue (NBUF=DEPTH+2)

        const float* row = rowbuf[rslot];

        if (t == 0) {
            if (tid < S_) {
                const float e = valid_s ? __logf(row[ext_s] + EPSV) : NEGV;
                float a = NEGV;
                if (tid == 0)      a = e;
                else if (tid == 1) a = (L > 0) ? e : NEGV;
                acur[tid] = a;
            }
        } else {
            if (tid < S_) {
                const float e  = valid_s ? __logf(row[ext_s] + EPSV) : NEGV;
                const float a1 = acur[tid];
                const float a2 = (tid >= 1) ? acur[tid - 1] : NEGV;
                const float a3 = skip_s ? acur[tid - 2] : NEGV;
                const float m  = fmaxf(a1, fmaxf(a2, a3));
                const float s  = __expf(a1 - m) + __expf(a2 - m) + __expf(a3 - m);
                const float nv = m + __logf(s) + e;
                anxt[tid] = (t < Tlen) ? nv : a1;   // freeze past input_length
            }
            float* tmp = acur; acur = anxt; anxt = tmp;
        }

        rslot = (rslot + 1 == NBUF) ? 0 : rslot + 1;
        pslot = (pslot + 1 == NBUF) ? 0 : pslot + 1;
        pvoff += ROWB;
    }

    __syncthreads();   // make final alpha visible to lane 0

    if (tid == 0) {
        const float a_last = acur[2 * L];
        const float a_prev = (L > 0) ? acur[2 * L - 1] : NEGV;
        const float m  = fmaxf(a_last, a_prev);
        const float ll = m + __logf(__expf(a_last - m) + __expf(a_prev - m));
        out[b] = -ll;
    }
}

extern "C" void kernel_launch(void* const* d_in, const int* in_sizes, int n_in,
                              void* d_out, int out_size, void* d_ws, size_t ws_size,
                              hipStream_t stream) {
    (void)in_sizes; (void)n_in; (void)d_ws; (void)ws_size; (void)out_size;
    const int*   y_true       = (const int*)d_in[0];
    const float* y_pred       = (const float*)d_in[1];
    const int*   input_length = (const int*)d_in[2];
    const int*   label_length = (const int*)d_in[3];
    float*       out          = (float*)d_out;

    hipLaunchKernelGGL(ctc_loss_kernel, dim3(B_), dim3(BLOCK), 0, stream,
                       y_true, y_pred, input_length, label_length, out);
}